// Gaussian2dRender_24988119728210
// MI455X (gfx1250) — compile-verified
//
#include <hip/hip_runtime.h>
#include <math.h>

// Problem constants (match reference)
#define BATCH     2
#define NGAUSS    4096
#define IMG_H     256
#define IMG_W     256
#define TILE_SZ   16
#define ALPHA_MIN (1.0f/255.0f)
#define ALPHA_MAX 0.99f

// Kernel organization
#define NTHREADS  512            // 16 wave32s per block; one block per (tile,batch)
#define NWAVES    (NTHREADS/32)
#define CHUNK     512            // gaussians tested per compaction pass
#define LIST_CAP  (CHUNK + 16)   // +16 pad slots so group reads never go OOB

typedef __attribute__((ext_vector_type(2))) float v2f;
typedef __attribute__((ext_vector_type(8))) float v8f;

__global__ __launch_bounds__(NTHREADS)
void gs2d_render(const int* __restrict__ batch_ids,
                 const float* __restrict__ means,
                 const float* __restrict__ sigmas,
                 const float* __restrict__ rhos,
                 const float* __restrict__ colors,
                 const float* __restrict__ opacs,
                 float* __restrict__ out)
{
    __shared__ float s_coef[LIST_CAP][8];   // [c0..c5, pad, pad] per gaussian
    __shared__ float s_col [LIST_CAP][4];   // rgba per gaussian
    __shared__ int   s_wcnt[NWAVES];
    __shared__ int   s_woff[NWAVES];
    __shared__ int   s_count;

    const int tx   = blockIdx.x;            // tile col
    const int ty   = blockIdx.y;            // tile row
    const int bz   = blockIdx.z;            // batch
    const int tid  = threadIdx.x;
    const int wave = tid >> 5;
    const int lane = tid & 31;
    const int half = lane >> 4;             // 0: lanes 0-15, 1: lanes 16-31
    const int l16  = lane & 15;

    // This wave renders pixel row (ty*16 + wave); lanes p and p+16 share pixel column l16.
    const int   py = ty * TILE_SZ + wave;
    const float y  = (float)py + 0.5f;
    const float x  = (float)(tx * TILE_SZ + l16) + 0.5f;

    // B-matrix of the two chained WMMAs (features per pixel column), built once.
    // Assumed 32-bit B 4x16 layout: lane holds column N=l16; VGPR0 = K(half?2:0), VGPR1 = K(half?3:1).
    // WMMA #1 features K0..K3 = [x^2, x*y, y^2, x]; WMMA #2 K0..K3 = [y, 1, 0, 0].
    v2f bf1, bf2;
    bf1.x = half ? y * y : x * x;
    bf1.y = half ? x     : x * y;
    bf2.x = half ? 0.0f  : y;
    bf2.y = half ? 0.0f  : 1.0f;

    float T = 1.0f;                          // transmittance (both lane-halves track full T)
    float acc0 = 0.0f, acc1 = 0.0f, acc2 = 0.0f, acc3 = 0.0f;
    bool  done = false;

    const unsigned long long lmask_lt = (1ull << lane) - 1ull;
    const float invTile = 1.0f / (float)TILE_SZ;
    const float ftx = (float)tx, fty = (float)ty;

    for (int cbase = 0; cbase < NGAUSS; cbase += CHUNK) {
        if (cbase + CHUNK < NGAUSS) {
            __builtin_prefetch(means + 2 * (cbase + CHUNK), 0, 0);   // global_prefetch_b8
            __builtin_prefetch(colors + 4 * (cbase + CHUNK), 0, 0);
        }

        // ---- Phase A: test + ordered compaction of this chunk into LDS ----
        const int g = cbase + tid;
        bool valid = false;
        float c0 = 0, c1 = 0, c2 = 0, c3 = 0, c4 = 0, c5 = 0;
        float r0 = 0, r1 = 0, r2 = 0, r3 = 0;
        if (g < NGAUSS && batch_ids[g] == bz) {
            const float mx = means[2 * g], my = means[2 * g + 1];
            const float sx = sigmas[2 * g], sy = sigmas[2 * g + 1];
            const float rad = 3.0f * sqrtf(fmaxf(sx, sy));
            const float tlx = floorf((mx - rad) * invTile);
            const float thx = floorf((mx + rad) * invTile);
            const float tly = floorf((my - rad) * invTile);
            const float thy = floorf((my + rad) * invTile);
            if (ftx >= tlx && ftx <= thx && fty >= tly && fty <= thy) {
                valid = true;
                const float rho = rhos[g];
                const float co = cosf(rho), si = sinf(rho);
                const float isx = 1.0f / sx, isy = 1.0f / sy;   // reference uses 1/sigma directly
                const float a  = co * co * isx + si * si * isy;
                const float bq = co * si * (isx - isy);
                const float cq = si * si * isx + co * co * isy;
                const float op = opacs[g];
                // e = 0.5*q - ln(op) = F . C  with F = [x^2, xy, y^2, x, y, 1]
                c0 = 0.5f * a;
                c1 = bq;
                c2 = 0.5f * cq;
                c3 = -(a * mx + bq * my);
                c4 = -(bq * mx + cq * my);
                c5 = 0.5f * a * mx * mx + bq * mx * my + 0.5f * cq * my * my - logf(op);
                r0 = colors[4 * g + 0];
                r1 = colors[4 * g + 1];
                r2 = colors[4 * g + 2];
                r3 = colors[4 * g + 3];
            }
        }

        const unsigned long long bal = __ballot(valid);
        if (lane == 0) s_wcnt[wave] = (int)__popcll(bal);
        __syncthreads();
        if (tid == 0) {
            int s = 0;
            #pragma unroll
            for (int i = 0; i < NWAVES; ++i) { s_woff[i] = s; s += s_wcnt[i]; }
            s_count = s;
        }
        __syncthreads();
        if (valid) {
            const int pos = s_woff[wave] + (int)__popcll(bal & lmask_lt);
            float* cw = &s_coef[pos][0];
            cw[0] = c0; cw[1] = c1; cw[2] = c2; cw[3] = c3;
            cw[4] = c4; cw[5] = c5; cw[6] = 0.0f; cw[7] = 0.0f;
            float* lw = &s_col[pos][0];
            lw[0] = r0; lw[1] = r1; lw[2] = r2; lw[3] = r3;
        }
        const int cnt = s_count;
        if (tid < 16) {   // pad slots -> alpha = exp(-3e38) = 0
            float* cw = &s_coef[cnt + tid][0];
            cw[0] = 0; cw[1] = 0; cw[2] = 0; cw[3] = 0;
            cw[4] = 0; cw[5] = 3.0e38f; cw[6] = 0; cw[7] = 0;
            float* lw = &s_col[cnt + tid][0];
            lw[0] = 0; lw[1] = 0; lw[2] = 0; lw[3] = 0;
        }
        __syncthreads();

        // ---- Phase B: composite list in order, 16 gaussians per step ----
        if (!done && cnt > 0) {
            for (int base = 0; base < cnt; base += 16) {
                // A-matrix (coeffs): 32-bit A 16x4 layout — row M = l16, VGPR0/1 = K(2*half), K(2*half+1)
                const float* cp = &s_coef[base + l16][half << 1];
                const v2f a1 = *(const v2f*)(cp);        // WMMA1 coeff pair
                const v2f a2 = *(const v2f*)(cp + 4);    // WMMA2 coeff pair
                v8f cz = {};
                v8f e1 = __builtin_amdgcn_wmma_f32_16x16x4_f32(
                    false, a1, false, bf1, (short)0, cz, false, false);
                v8f e  = __builtin_amdgcn_wmma_f32_16x16x4_f32(
                    false, a2, false, bf2, (short)0, e1, false, false);

                // alpha for this lane's 8 gaussians (indices base + 8*half + r)
                float alpha[8];
                #pragma unroll
                for (int r = 0; r < 8; ++r) {
                    float ar = __expf(-e[r]);
                    ar = fminf(ALPHA_MAX, ar);
                    alpha[r] = (ar < ALPHA_MIN) ? 0.0f : ar;
                }
                // ordered prefix product of (1-alpha) within the lane-half
                float w[8];
                float pref = 1.0f;
                #pragma unroll
                for (int r = 0; r < 8; ++r) {
                    w[r] = pref * alpha[r];
                    pref *= (1.0f - alpha[r]);
                }
                // stitch halves: upper half (gaussians 8..15) is scaled by lower half's product
                const float other  = __shfl_xor(pref, 16, 32);
                const float wscale = half ? (T * other) : T;
                const float* colp = &s_col[base + (half << 3)][0];
                #pragma unroll
                for (int r = 0; r < 8; ++r) {
                    const float wr = wscale * w[r];
                    acc0 = fmaf(wr, colp[4 * r + 0], acc0);
                    acc1 = fmaf(wr, colp[4 * r + 1], acc1);
                    acc2 = fmaf(wr, colp[4 * r + 2], acc2);
                    acc3 = fmaf(wr, colp[4 * r + 3], acc3);
                }
                T *= pref * other;    // same value in both halves
            }
            done = (__all(T < 1.0e-4f) != 0);   // wave-level early out
        }
        __syncthreads();   // protect LDS before next chunk overwrites
    }

    // combine the two lane-half partial sums for each pixel, then store
    acc0 += __shfl_xor(acc0, 16, 32);
    acc1 += __shfl_xor(acc1, 16, 32);
    acc2 += __shfl_xor(acc2, 16, 32);
    acc3 += __shfl_xor(acc3, 16, 32);
    if (half == 0) {
        const int px = tx * TILE_SZ + l16;
        const size_t plane = (size_t)IMG_H * IMG_W;
        const size_t o = ((size_t)bz * 4) * plane + (size_t)py * IMG_W + px;
        out[o]             = acc0;
        out[o + plane]     = acc1;
        out[o + 2 * plane] = acc2;
        out[o + 3 * plane] = acc3;
    }
}

extern "C" void kernel_launch(void* const* d_in, const int* in_sizes, int n_in,
                              void* d_out, int out_size, void* d_ws, size_t ws_size,
                              hipStream_t stream) {
    (void)in_sizes; (void)n_in; (void)d_ws; (void)ws_size; (void)out_size;
    const int*   batch_ids = (const int*)  d_in[0];
    const float* means     = (const float*)d_in[1];
    const float* sigmas    = (const float*)d_in[2];
    const float* rhos      = (const float*)d_in[3];
    const float* colors    = (const float*)d_in[4];
    const float* opacs     = (const float*)d_in[5];
    float*       out       = (float*)d_out;

    dim3 grid(IMG_W / TILE_SZ, IMG_H / TILE_SZ, BATCH);
    gs2d_render<<<grid, NTHREADS, 0, stream>>>(batch_ids, means, sigmas, rhos,
                                               colors, opacs, out);
}